// PermutationMatchingLoss_52261162058276
// MI455X (gfx1250) — compile-verified
//
#include <hip/hip_runtime.h>
#include <cstdint>
#include <cstddef>

#define NHEADS 5
#define NCLASS 2048
#define NPERM  120
#define BLK    320   // 10 wave32 per workgroup

// ---- feature detection (device pass only; host pass always takes fallback) ----
#if defined(__HIP_DEVICE_COMPILE__)
  #if __has_builtin(__builtin_amdgcn_global_load_async_to_lds_b128) && \
      __has_builtin(__builtin_amdgcn_s_wait_asynccnt)
    #define USE_ASYNC_LDS 1
  #else
    #define USE_ASYNC_LDS 0
  #endif
  #if __has_builtin(__builtin_amdgcn_wmma_f32_16x16x4_f32)
    #define USE_WMMA_PERM 1
  #else
    #define USE_WMMA_PERM 0
  #endif
#else
  #define USE_ASYNC_LDS 0
  #define USE_WMMA_PERM 0
#endif

#if USE_ASYNC_LDS
// Matches the builtin's parameter: pointer to 16-byte generic int vector.
typedef int av4i __attribute__((vector_size(16)));
typedef __attribute__((address_space(1))) av4i* gv4i_p;   // global (AS1)
typedef __attribute__((address_space(3))) av4i* lv4i_p;   // LDS (AS3)
#endif

// Lexicographic rank -> permutation of {0..4} (Lehmer / factoradic decode).
__device__ __forceinline__ void lehmer5(int r, int perm[5]) {
  int avail[5] = {0, 1, 2, 3, 4};
  const int f[4] = {24, 6, 2, 1};
#pragma unroll
  for (int pos = 0; pos < 4; ++pos) {
    const int d = r / f[pos];
    r -= d * f[pos];
    perm[pos] = avail[d];
#pragma unroll
    for (int k = 0; k < 4; ++k)
      if (k >= d) avail[k] = avail[k + 1];
  }
  perm[4] = avail[0];
}

__global__ __launch_bounds__(BLK) void pml_main(const float* __restrict__ pred,
                                                const float* __restrict__ tgt,
                                                float* __restrict__ partial) {
  __shared__ __align__(16) float s_pred[NHEADS * NCLASS];  // 40 KB staged row
  __shared__ float s_lse[NHEADS];
  __shared__ int   s_tidx[NHEADS];
  __shared__ float s_ce[NHEADS * NHEADS];
  __shared__ float s_wv[10];
  __shared__ int   s_wi[10];

  const int b    = blockIdx.x;
  const int tid  = threadIdx.x;
  const int lane = tid & 31;
  const int wave = tid >> 5;

  const float* predb = pred + (size_t)b * (NHEADS * NCLASS);
  const float* tgtb  = tgt  + (size_t)b * (NHEADS * NCLASS);

  // ---- Stage prediction row [5 x 2048] f32 -> LDS ----
#if USE_ASYNC_LDS
  // 2560 float4 / 320 threads = 8 async B128 issues per thread. Fire and
  // continue: the targets-argmax streaming pass below overlaps the transfer.
#pragma unroll
  for (int r = 0; r < 8; ++r) {
    const int i4 = r * BLK + tid;
    __builtin_amdgcn_global_load_async_to_lds_b128(
        (gv4i_p)(predb + i4 * 4),
        (lv4i_p)(s_pred + i4 * 4),
        0, 0);
  }
#else
  {
    const float4* g4 = reinterpret_cast<const float4*>(predb);
    float4* l4 = reinterpret_cast<float4*>(s_pred);
#pragma unroll
    for (int r = 0; r < 8; ++r) {
      const int i4 = r * BLK + tid;
      l4[i4] = g4[i4];
    }
  }
#endif

  // ---- targets argmax: 2 waves per head, streamed from global (read once) ----
  {
    const int h  = wave >> 1;                    // head 0..4
    const int lp = ((wave & 1) << 5) | lane;     // 0..63 within head
    const float4* t4 = reinterpret_cast<const float4*>(tgtb + h * NCLASS);
    float bv = -3.402823466e+38f;
    int   bi = 0;
#pragma unroll
    for (int r = 0; r < 8; ++r) {
      const int i4 = r * 64 + lp;
      const float4 v = t4[i4];
      const int e = i4 * 4;
      if (v.x > bv) { bv = v.x; bi = e;     }   // ascending scan => first-max kept
      if (v.y > bv) { bv = v.y; bi = e + 1; }
      if (v.z > bv) { bv = v.z; bi = e + 2; }
      if (v.w > bv) { bv = v.w; bi = e + 3; }
    }
#pragma unroll
    for (int off = 16; off > 0; off >>= 1) {
      const float ov = __shfl_xor(bv, off, 32);
      const int   oi = __shfl_xor(bi, off, 32);
      if (ov > bv || (ov == bv && oi < bi)) { bv = ov; bi = oi; }
    }
    if (lane == 0) { s_wv[wave] = bv; s_wi[wave] = bi; }
  }

#if USE_ASYNC_LDS
  __builtin_amdgcn_s_wait_asynccnt(0);
#endif
  __syncthreads();  // s_pred staged + per-wave argmax candidates visible

  // combine the two wave candidates per head
  if (tid < NHEADS) {
    const float v0 = s_wv[2 * tid], v1 = s_wv[2 * tid + 1];
    const int   i0 = s_wi[2 * tid], i1 = s_wi[2 * tid + 1];
    s_tidx[tid] = (v1 > v0 || (v1 == v0 && i1 < i0)) ? i1 : i0;
  }

  // ---- logsumexp per head from LDS (waves 5..9, one head each) ----
  if (wave >= 5) {
    const int h = wave - 5;
    const float4* p4 = reinterpret_cast<const float4*>(s_pred + h * NCLASS);
    float m = -3.402823466e+38f;
#pragma unroll
    for (int r = 0; r < 16; ++r) {
      const float4 v = p4[r * 32 + lane];
      m = fmaxf(m, fmaxf(fmaxf(v.x, v.y), fmaxf(v.z, v.w)));
    }
#pragma unroll
    for (int off = 16; off > 0; off >>= 1)
      m = fmaxf(m, __shfl_xor(m, off, 32));
    float s = 0.0f;
#pragma unroll
    for (int r = 0; r < 16; ++r) {
      const float4 v = p4[r * 32 + lane];
      s += __expf(v.x - m) + __expf(v.y - m) + __expf(v.z - m) + __expf(v.w - m);
    }
#pragma unroll
    for (int off = 16; off > 0; off >>= 1)
      s += __shfl_xor(s, off, 32);
    if (lane == 0) s_lse[h] = m + __logf(s);
  }
  __syncthreads();

  // ce[i][j] = lse[i] - pred[i, t[j]]   (gather comes free out of LDS)
  if (tid < NHEADS * NHEADS) {
    const int i = tid / NHEADS, j = tid % NHEADS;
    s_ce[tid] = s_lse[i] - s_pred[i * NCLASS + s_tidx[j]];
  }
  __syncthreads();

  // ---- min over 120 permutations (wave 0; EXEC all-ones for WMMA) ----
  if (wave == 0) {
    float best = 3.402823466e+38f;
#if USE_WMMA_PERM
    // scores(120) = OneHotSel(120x25) . vec(ce)(25), as 8 row-tiles x 7 K-chunks
    // of v_wmma_f32_16x16x4_f32; B is vec(ce) replicated across all 16 columns.
    typedef __attribute__((ext_vector_type(2))) float v2f;
    typedef __attribute__((ext_vector_type(8))) float v8f;
    const int Mrow  = lane & 15;
    const int khalf = (lane >> 4) << 1;  // lanes 0-15 hold K0/K1, lanes 16-31 K2/K3
#pragma unroll
    for (int tile = 0; tile < 8; ++tile) {
      const int p = tile * 16 + Mrow;
      int perm[5];
      lehmer5(p < NPERM ? p : 0, perm);
      v8f acc = {0.f, 0.f, 0.f, 0.f, 0.f, 0.f, 0.f, 0.f};
#pragma unroll
      for (int kc = 0; kc < 7; ++kc) {
        const int c0 = kc * 4 + khalf;   // flat index into 25-wide K (padded to 28)
        const int c1 = c0 + 1;
        v2f a, bb;
        a.x = (c0 < 25 && perm[c0 % 5] == c0 / 5) ? 1.0f : 0.0f;
        a.y = (c1 < 25 && perm[c1 % 5] == c1 / 5) ? 1.0f : 0.0f;
        bb.x = (c0 < 25) ? s_ce[c0] : 0.0f;
        bb.y = (c1 < 25) ? s_ce[c1] : 0.0f;
        acc = __builtin_amdgcn_wmma_f32_16x16x4_f32(false, a, false, bb,
                                                    (short)0, acc, false, false);
      }
#pragma unroll
      for (int r = 0; r < 8; ++r) {
        const int M  = (lane < 16) ? r : (r + 8);  // C/D layout: vgpr r -> rows r, r+8
        const int pr = tile * 16 + M;
        if (pr < NPERM) best = fminf(best, acc[r]);
      }
    }
#else
    for (int p = lane; p < NPERM; p += 32) {
      int perm[5];
      lehmer5(p, perm);
      float sc = 0.0f;
#pragma unroll
      for (int h = 0; h < NHEADS; ++h) sc += s_ce[perm[h] * NHEADS + h];
      best = fminf(best, sc);
    }
#endif
#pragma unroll
    for (int off = 16; off > 0; off >>= 1)
      best = fminf(best, __shfl_xor(best, off, 32));
    // min perm loss is a sum over heads; x 1/NHEADS gives the per-row CE mean (ties irrelevant: value only)
    if (lane == 0) partial[b] = best * (1.0f / NHEADS);
  }
}

// Deterministic fixed-order reduction of per-row partials -> scalar mean.
__global__ __launch_bounds__(256) void pml_reduce(const float* __restrict__ partial,
                                                  float* __restrict__ out, int n) {
  __shared__ float sm[256];
  float s = 0.0f;
  for (int i = threadIdx.x; i < n; i += 256) s += partial[i];
  sm[threadIdx.x] = s;
  __syncthreads();
  for (int off = 128; off > 0; off >>= 1) {
    if ((int)threadIdx.x < off) sm[threadIdx.x] += sm[threadIdx.x + off];
    __syncthreads();
  }
  if (threadIdx.x == 0) out[0] = sm[0] / (float)n;
}

extern "C" void kernel_launch(void* const* d_in, const int* in_sizes, int n_in,
                              void* d_out, int out_size, void* d_ws, size_t ws_size,
                              hipStream_t stream) {
  (void)n_in; (void)out_size; (void)ws_size;
  const float* pred = (const float*)d_in[0];
  const float* tgt  = (const float*)d_in[1];
  float* out     = (float*)d_out;
  float* partial = (float*)d_ws;                    // nb floats of scratch
  const int nb = in_sizes[0] / (NHEADS * NCLASS);   // 16384
  pml_main<<<nb, BLK, 0, stream>>>(pred, tgt, partial);
  pml_reduce<<<1, 256, 0, stream>>>(partial, out, nb);
}